// PTCFNORef_80066780332650
// MI455X (gfx1250) — compile-verified
//
#include <hip/hip_runtime.h>
#include <math.h>

// ---------------- problem constants ----------------
#define IMG      2048
#define NB       16        // batch
#define DCH      64        // spectral channels
#define INF      256       // features per patch (16x16)
#define NPOS     23        // needed patch rows/cols per image (valid)
#define GRID24   24        // staging grid incl. zero pad row/col 0
#define NPATCH   (NPOS*NPOS)       // 529
#define NTILE    ((NPATCH+15)/16)  // 34 patch tiles of 16

typedef float v2f __attribute__((ext_vector_type(2)));
typedef float v8f __attribute__((ext_vector_type(8)));

// ---------------- precompute: twiddle tables ----------------
__global__ void k_tables(double* ct256, double* st256, double* ct64, double* st64) {
    int t = threadIdx.x;
    const double TWO_PI = 6.283185307179586476925286766559;
    if (t < 256) { double a = TWO_PI * (double)t / 256.0; ct256[t] = cos(a); st256[t] = sin(a); }
    if (t < 64)  { double a = TWO_PI * (double)t / 64.0;  ct64[t]  = cos(a); st64[t]  = sin(a); }
}

// B1 = Wr*C + Wi*S ;  B2 = Wi*C - Wr*S   (C[k,n]=cos(2pi kn/256), S=sin)
__global__ void k_b12(const float* __restrict__ Wr, const float* __restrict__ Wi,
                      const double* __restrict__ ct, const double* __restrict__ st,
                      double* __restrict__ B1, double* __restrict__ B2) {
    int d = blockIdx.x;      // 0..63
    int n = threadIdx.x;     // 0..255
    const float* wr = Wr + d * INF;
    const float* wi = Wi + d * INF;
    double a1 = 0.0, a2 = 0.0;
    for (int k = 0; k < INF; ++k) {
        int t = (k * n) & 255;
        double c = ct[t], s = st[t];
        double r = (double)wr[k], im = (double)wi[k];
        a1 += r * c + im * s;
        a2 += im * c - r * s;
    }
    B1[d * INF + n] = a1;
    B2[d * INF + n] = a2;
}

// A[m,n] = (1/64) sum_d ( cos(2pi dm/64)*B1[d,n] - sin(...)*B2[d,n] ),  cvec likewise from biases
__global__ void k_amat(const double* __restrict__ B1, const double* __restrict__ B2,
                       const double* __restrict__ ct64, const double* __restrict__ st64,
                       const float* __restrict__ br, const float* __restrict__ bi,
                       float* __restrict__ A, float* __restrict__ cvec) {
    int m = blockIdx.x;      // 0..63
    int n = threadIdx.x;     // 0..255
    double acc = 0.0;
    for (int d = 0; d < DCH; ++d) {
        int t = (d * m) & 63;
        acc += ct64[t] * B1[d * INF + n] - st64[t] * B2[d * INF + n];
    }
    A[m * INF + n] = (float)(acc / 64.0);
    if (n == 0) {
        double cc = 0.0;
        for (int d = 0; d < DCH; ++d) {
            int t = (d * m) & 63;
            cc += ct64[t] * ((double)br[d] - (double)bi[d])
                - st64[t] * ((double)br[d] + (double)bi[d]);
        }
        cvec[m] = (float)(cc / 64.0);
    }
}

__global__ void k_zero(float* __restrict__ p, int n) {
    int i = blockIdx.x * blockDim.x + threadIdx.x;
    if (i < n) p[i] = 0.0f;
}

// ---------------- main WMMA kernel: f = A * patch + c at the 23x23 needed positions ----------------
// One block = 128 threads = 4 waves; wave w handles channel tile w (16 channels).
// Block covers one patch tile (16 patches) of one image.
__global__ void k_wmma(const float* __restrict__ x, const float* __restrict__ A,
                       const float* __restrict__ cvec, float* __restrict__ fbuf) {
    int lane = threadIdx.x & 31;
    int wv   = threadIdx.x >> 5;          // channel tile 0..3
    int pt   = blockIdx.x % NTILE;        // patch tile 0..33
    int img  = blockIdx.x / NTILE;        // 0..15
    int n    = lane & 15;                 // N index (patch within tile)
    int hi   = lane >> 4;                 // lane half -> K offset {0,2}

    int q  = pt * 16 + n;
    int qc = q > (NPATCH - 1) ? (NPATCH - 1) : q;   // clamp tail (discard at store)
    int ri = 1 + qc / NPOS;               // staging row 1..23
    int ci = 1 + qc % NPOS;               // staging col 1..23
    int prow = 16 * (ri / 3) + (ri % 3) - 1;   // patch-grid row 0..113
    int pcol = 16 * (ci / 3) + (ci % 3) - 1;   // patch-grid col 0..113

    // B operand base: this lane supplies patch n, features 4s+2*hi, 4s+2*hi+1
    const float* xb = x + (size_t)img * IMG * IMG
                        + (size_t)(prow * 16) * IMG + (size_t)(pcol * 16) + 2 * hi;
    // A operand base: this lane supplies channel row wv*16+n, features 4s+2*hi .. +1
    const float* ap = A + (size_t)(wv * 16 + n) * INF + 2 * hi;

    __builtin_prefetch((const void*)xb, 0, 3);     // global_prefetch_b8

    v8f acc = {};
    #pragma unroll 8
    for (int s = 0; s < 64; ++s) {
        v2f a = *(const v2f*)(ap + 4 * s);
        v2f b = *(const v2f*)(xb + (s >> 2) * IMG + ((s & 3) << 2));
        acc = __builtin_amdgcn_wmma_f32_16x16x4_f32(
                  /*neg_a=*/false, a, /*neg_b=*/false, b,
                  /*c_mod=*/(short)0, acc, /*reuse_a=*/false, /*reuse_b=*/false);
    }

    // D layout: VGPR j -> M = j + 8*hi (lane half), N = lane&15
    if (q <= NPATCH - 1) {
        #pragma unroll
        for (int j = 0; j < 8; ++j) {
            int ch = wv * 16 + hi * 8 + j;
            fbuf[(((size_t)img * DCH + ch) * GRID24 + ri) * GRID24 + ci] = acc[j] + cvec[ch];
        }
    }
}

// ---------------- epilogue: depthwise 3x3 conv + bias + BN + 8x8 nearest sample ----------------
__global__ void k_out(const float* __restrict__ fbuf, const float* __restrict__ cw,
                      const float* __restrict__ cb, const float* __restrict__ gamma,
                      const float* __restrict__ beta, const float* __restrict__ mean,
                      const float* __restrict__ var, float* __restrict__ out) {
    int idx = blockIdx.x * blockDim.x + threadIdx.x;
    if (idx >= NB * DCH * 8 * 8) return;
    int ow = idx & 7;
    int oh = (idx >> 3) & 7;
    int ch = (idx >> 6) & 63;
    int b  = idx >> 12;
    const float* f = fbuf + (((size_t)b * DCH + ch) * GRID24 + 3 * oh) * GRID24 + 3 * ow;
    const float* w = cw + ch * 9;
    float acc = cb[ch];
    #pragma unroll
    for (int j = 0; j < 3; ++j)
        #pragma unroll
        for (int i = 0; i < 3; ++i)
            acc += f[j * GRID24 + i] * w[j * 3 + i];
    float scale = gamma[ch] / sqrtf(var[ch] + 1e-5f);
    out[idx] = (acc - mean[ch]) * scale + beta[ch];
}

// ---------------- launch ----------------
extern "C" void kernel_launch(void* const* d_in, const int* in_sizes, int n_in,
                              void* d_out, int out_size, void* d_ws, size_t ws_size,
                              hipStream_t stream) {
    const float* x     = (const float*)d_in[0];
    const float* Wr    = (const float*)d_in[1];
    const float* br    = (const float*)d_in[2];
    const float* Wi    = (const float*)d_in[3];
    const float* bi    = (const float*)d_in[4];
    const float* cw    = (const float*)d_in[5];
    const float* cb    = (const float*)d_in[6];
    const float* gamma = (const float*)d_in[7];
    const float* beta  = (const float*)d_in[8];
    const float* mean  = (const float*)d_in[9];
    const float* var   = (const float*)d_in[10];

    char* ws = (char*)d_ws;
    double* ct256 = (double*)(ws + 0);        // 2048 B
    double* st256 = (double*)(ws + 2048);     // 2048 B
    double* ct64  = (double*)(ws + 4096);     // 512 B
    double* st64  = (double*)(ws + 4608);     // 512 B
    double* B1    = (double*)(ws + 5120);     // 131072 B
    double* B2    = (double*)(ws + 136192);   // 131072 B
    float*  A     = (float*) (ws + 267264);   // 65536 B
    float*  cvec  = (float*) (ws + 332800);   // 256 B
    float*  fbuf  = (float*) (ws + 333056);   // 16*64*24*24*4 = 2359296 B

    k_tables<<<1, 256, 0, stream>>>(ct256, st256, ct64, st64);
    k_b12<<<DCH, INF, 0, stream>>>(Wr, Wi, ct256, st256, B1, B2);
    k_amat<<<DCH, INF, 0, stream>>>(B1, B2, ct64, st64, br, bi, A, cvec);

    int nz = NB * DCH * GRID24 * GRID24;
    k_zero<<<(nz + 255) / 256, 256, 0, stream>>>(fbuf, nz);

    k_wmma<<<NB * NTILE, 128, 0, stream>>>(x, A, cvec, fbuf);

    int nout = NB * DCH * 8 * 8;
    k_out<<<(nout + 255) / 256, 256, 0, stream>>>(fbuf, cw, cb, gamma, beta, mean, var,
                                                  (float*)d_out);
}